// GNNActorPenta_30657476559584
// MI455X (gfx1250) — compile-verified
//
#include <hip/hip_runtime.h>
#include <math.h>

typedef __attribute__((ext_vector_type(2))) float v2f;
typedef __attribute__((ext_vector_type(8))) float v8f;

#define N_NODES 80896
#define N_EDGES 808960
#define N_GRAPH 1024
#define NODES_PER_G 79
#define C_IN 64
#define HID 32
#define MTILES 5056   // 80896 / 16

__constant__ int POS_IDX[79] = {
  120,124,128,132,136,140,144,148,152,237,241,245,249,253,257,261,265,269,
  354,358,362,366,370,374,378,382,386,471,475,479,483,487,491,495,499,503,
  588,592,596,600,604,608,612,616,620,705,709,713,717,721,725,729,733,737,
  822,826,830,834,838,842,846,850,854,48,53,60,67,73,157,352,388,583,586,
  817,901,906,913,920,926
};

// ---------------- degree / normalization ----------------
__global__ void init_deg(float* deg, int n) {
  int i = blockIdx.x * blockDim.x + threadIdx.x;
  if (i < n) deg[i] = 1.0f;              // self loop
}
__global__ void count_deg(const int* __restrict__ dst, float* deg, int nE) {
  int i = blockIdx.x * blockDim.x + threadIdx.x;
  if (i < nE) atomicAdd(&deg[dst[i]], 1.0f);
}
__global__ void finish_deg(float* deg, int n) {  // deg -> dinv in place
  int i = blockIdx.x * blockDim.x + threadIdx.x;
  if (i < n) deg[i] = rsqrtf(deg[i]);
}

// ---------------- total_agents per graph ----------------
__global__ void ta_kernel(const float* __restrict__ state, float* ta) {
  int g = blockIdx.x * blockDim.x + threadIdx.x;
  if (g >= N_GRAPH) return;
  float s = 0.f;
  const float* p = state + (size_t)g * NODES_PER_G * C_IN + 1;
  for (int i = 0; i < NODES_PER_G; ++i) s += p[(size_t)i * C_IN];
  ta[g] = s;
}

// ---------------- GCN feature GEMM: H = X @ W  (NxC_IN @ C_INxC_IN) ----------------
__global__ void gemm64_wmma(const float* __restrict__ X, const float* __restrict__ W,
                            float* __restrict__ H) {
  int lane = threadIdx.x & 31;
  int wave = threadIdx.x >> 5;           // 4 waves -> 4 N-tiles
  int row  = lane & 15;
  int half = lane >> 4;
  int m0 = blockIdx.x * 16;
  int n0 = wave * 16;
  v8f c = {};
  const float* xrow = X + (size_t)(m0 + row) * C_IN;
  #pragma unroll
  for (int kt = 0; kt < 16; ++kt) {
    int k = kt * 4 + half * 2;
    v2f a; a[0] = xrow[k];                       a[1] = xrow[k + 1];
    v2f b; b[0] = W[(size_t)k * C_IN + n0 + row]; b[1] = W[(size_t)(k + 1) * C_IN + n0 + row];
    c = __builtin_amdgcn_wmma_f32_16x16x4_f32(false, a, false, b, (short)0, c, false, false);
  }
  #pragma unroll
  for (int v = 0; v < 8; ++v)
    H[(size_t)(m0 + v + 8 * half) * C_IN + n0 + row] = c[v];
}

// ---------------- aggregation: init (bias + self loop), edges (atomics), relu ----------------
__global__ void agg_init(const float* __restrict__ H, const float* __restrict__ dinv,
                         const float* __restrict__ bias, float* __restrict__ out, int n) {
  int i = blockIdx.x * blockDim.x + threadIdx.x;
  if (i >= n) return;
  int node = i >> 6, ch = i & 63;
  float w = dinv[node];
  out[i] = bias[ch] + H[i] * w * w;
}
__global__ void agg_edges(const int* __restrict__ src, const int* __restrict__ dst,
                          const float* __restrict__ H, const float* __restrict__ dinv,
                          float* __restrict__ out, int nE) {
  int t = blockIdx.x * blockDim.x + threadIdx.x;
  if (t >= nE * 16) return;
  int e = t >> 4;
  int q = (t & 15) << 2;
  int s = src[e], d = dst[e];
  float w = dinv[s] * dinv[d];
  const float4 h = *(const float4*)(H + (size_t)s * C_IN + q);
  float* o = out + (size_t)d * C_IN + q;
  atomicAdd(o + 0, h.x * w);
  atomicAdd(o + 1, h.y * w);
  atomicAdd(o + 2, h.z * w);
  atomicAdd(o + 3, h.w * w);
}
__global__ void relu_ip(float* x, int n) {
  int i = blockIdx.x * blockDim.x + threadIdx.x;
  if (i < n) x[i] = fmaxf(x[i], 0.0f);
}

// ---------------- lin1: init with bias + scalar features, then K=64 WMMA accumulation ----------------
__global__ void lin1_init(const float* __restrict__ lin1_b, const float* __restrict__ L1W,
                          const float* __restrict__ ta, float* __restrict__ hid, int n) {
  int i = blockIdx.x * blockDim.x + threadIdx.x;
  if (i >= n) return;
  int m = i >> 5, j = i & 31;
  int g = m / NODES_PER_G;
  int p = POS_IDX[m - g * NODES_PER_G];
  float px = (float)(p % 39) * (1.0f / 38.0f);
  float py = (float)(p / 39) * (1.0f / 24.0f);
  hid[i] = lin1_b[j]
         + ta[g] * L1W[384 * HID + j]
         + px    * L1W[385 * HID + j]
         + py    * L1W[386 * HID + j];
}
// hid (Nx32) += X (Nx64) @ L1W[kofs:kofs+64, 0:32]
__global__ void lin1_accum(const float* __restrict__ X, const float* __restrict__ L1W,
                           float* __restrict__ hid, int kofs) {
  int lane = threadIdx.x & 31;
  int wave = threadIdx.x >> 5;           // 2 waves -> 2 N-tiles (32 cols)
  int row  = lane & 15;
  int half = lane >> 4;
  int m0 = blockIdx.x * 16;
  int n0 = wave * 16;
  v8f c;
  #pragma unroll
  for (int v = 0; v < 8; ++v)
    c[v] = hid[(size_t)(m0 + v + 8 * half) * HID + n0 + row];
  const float* xrow = X + (size_t)(m0 + row) * C_IN;
  #pragma unroll
  for (int kt = 0; kt < 16; ++kt) {
    int k = kt * 4 + half * 2;
    v2f a; a[0] = xrow[k]; a[1] = xrow[k + 1];
    v2f b; b[0] = L1W[(size_t)(kofs + k) * HID + n0 + row];
           b[1] = L1W[(size_t)(kofs + k + 1) * HID + n0 + row];
    c = __builtin_amdgcn_wmma_f32_16x16x4_f32(false, a, false, b, (short)0, c, false, false);
  }
  #pragma unroll
  for (int v = 0; v < 8; ++v)
    hid[(size_t)(m0 + v + 8 * half) * HID + n0 + row] = c[v];
}
__global__ void lrelu_ip(float* x, int n) {
  int i = blockIdx.x * blockDim.x + threadIdx.x;
  if (i < n) { float v = x[i]; x[i] = v >= 0.f ? v : 0.01f * v; }
}

// ---------------- lin2: hid2 = leaky_relu(hid1 @ W (32x32) + b) ----------------
__global__ void gemm_lin2(const float* __restrict__ H1, const float* __restrict__ W,
                          const float* __restrict__ bias, float* __restrict__ H2) {
  int lane = threadIdx.x & 31;
  int wave = threadIdx.x >> 5;
  int row  = lane & 15;
  int half = lane >> 4;
  int m0 = blockIdx.x * 16;
  int n0 = wave * 16;
  float bv = bias[n0 + row];
  v8f c;
  #pragma unroll
  for (int v = 0; v < 8; ++v) c[v] = bv;
  const float* xrow = H1 + (size_t)(m0 + row) * HID;
  #pragma unroll
  for (int kt = 0; kt < 8; ++kt) {
    int k = kt * 4 + half * 2;
    v2f a; a[0] = xrow[k]; a[1] = xrow[k + 1];
    v2f b; b[0] = W[(size_t)k * HID + n0 + row]; b[1] = W[(size_t)(k + 1) * HID + n0 + row];
    c = __builtin_amdgcn_wmma_f32_16x16x4_f32(false, a, false, b, (short)0, c, false, false);
  }
  #pragma unroll
  for (int v = 0; v < 8; ++v) {
    float x = c[v];
    H2[(size_t)(m0 + v + 8 * half) * HID + n0 + row] = x >= 0.f ? x : 0.01f * x;
  }
}

// ---------------- lin3 + softplus + reductions ----------------
__global__ void zero_small(float* gsum, float* regsum) {
  int i = blockIdx.x * blockDim.x + threadIdx.x;
  if (i < N_GRAPH) gsum[i] = 0.f;
  if (i == N_GRAPH) regsum[0] = 0.f;
}
__global__ void lin3_kernel(const float* __restrict__ H2, const float* __restrict__ w3,
                            const float* __restrict__ b3, float* __restrict__ out,
                            float* gsum, float* regsum, int nNodes) {
  int m = blockIdx.x * blockDim.x + threadIdx.x;
  float sp = 0.f;
  if (m < nNodes) {
    float acc = b3[0];
    const float* h = H2 + (size_t)m * HID;
    #pragma unroll
    for (int j = 0; j < HID; ++j) acc += h[j] * w3[j];
    sp = acc > 30.f ? acc : log1pf(expf(acc));   // softplus (>=0, so |sp| == sp)
    out[m] = sp;
    atomicAdd(&gsum[m / NODES_PER_G], sp);
  }
  __shared__ float red[256];
  red[threadIdx.x] = sp;
  __syncthreads();
  for (int s = 128; s > 0; s >>= 1) {
    if ((int)threadIdx.x < s) red[threadIdx.x] += red[threadIdx.x + s];
    __syncthreads();
  }
  if (threadIdx.x == 0) atomicAdd(regsum, red[0]);
}
__global__ void normalize_kernel(float* out, const float* __restrict__ gsum,
                                 const float* __restrict__ regsum, int nNodes) {
  int m = blockIdx.x * blockDim.x + threadIdx.x;
  if (m < nNodes) out[m] = out[m] / (gsum[m / NODES_PER_G] + 1e-20f);
  if (m == 0) out[nNodes] = regsum[0] * (1.0f / (float)N_NODES);
}

// ---------------- host side ----------------
extern "C" void kernel_launch(void* const* d_in, const int* in_sizes, int n_in,
                              void* d_out, int out_size, void* d_ws, size_t ws_size,
                              hipStream_t stream) {
  const float* state  = (const float*)d_in[0];
  const int*   eidx   = (const int*)  d_in[1];
  const float* W1     = (const float*)d_in[2];
  const float* b1     = (const float*)d_in[3];
  const float* W2     = (const float*)d_in[4];
  const float* b2     = (const float*)d_in[5];
  const float* W3     = (const float*)d_in[6];
  const float* b3g    = (const float*)d_in[7];
  const float* lin1_w = (const float*)d_in[8];
  const float* lin1_b = (const float*)d_in[9];
  const float* lin2_w = (const float*)d_in[10];
  const float* lin2_b = (const float*)d_in[11];
  const float* lin3_w = (const float*)d_in[12];
  const float* lin3_b = (const float*)d_in[13];
  float* out = (float*)d_out;

  const int* src = eidx;
  const int* dst = eidx + N_EDGES;

  // workspace layout (floats)
  float* ws = (float*)d_ws;
  size_t off = 0;
  float* Htmp = ws + off;  off += (size_t)N_NODES * C_IN;
  float* bufA = ws + off;  off += (size_t)N_NODES * C_IN;
  float* bufB = ws + off;  off += (size_t)N_NODES * C_IN;
  float* dinv = ws + off;  off += N_NODES;
  float* ta   = ws + off;  off += N_GRAPH;
  float* hid1 = ws + off;  off += (size_t)N_NODES * HID;
  float* hid2 = ws + off;  off += (size_t)N_NODES * HID;
  float* gsum = ws + off;  off += N_GRAPH;
  float* rsum = ws + off;  off += 1;

  const int NT = 256;
  // degree -> dinv
  init_deg<<<(N_NODES + NT - 1) / NT, NT, 0, stream>>>(dinv, N_NODES);
  count_deg<<<(N_EDGES + NT - 1) / NT, NT, 0, stream>>>(dst, dinv, N_EDGES);
  finish_deg<<<(N_NODES + NT - 1) / NT, NT, 0, stream>>>(dinv, N_NODES);
  // scalar features for lin1
  ta_kernel<<<(N_GRAPH + NT - 1) / NT, NT, 0, stream>>>(state, ta);
  lin1_init<<<((size_t)N_NODES * HID + NT - 1) / NT, NT, 0, stream>>>(lin1_b, lin1_w, ta, hid1, N_NODES * HID);
  lin1_accum<<<MTILES, 64, 0, stream>>>(state, lin1_w, hid1, 320);   // s3 block

  auto gcn = [&](const float* xin, const float* W, const float* b, float* xout, int kofs) {
    gemm64_wmma<<<MTILES, 128, 0, stream>>>(xin, W, Htmp);
    agg_init<<<((size_t)N_NODES * C_IN + NT - 1) / NT, NT, 0, stream>>>(Htmp, dinv, b, xout, N_NODES * C_IN);
    agg_edges<<<((size_t)N_EDGES * 16 + NT - 1) / NT, NT, 0, stream>>>(src, dst, Htmp, dinv, xout, N_EDGES);
    relu_ip<<<((size_t)N_NODES * C_IN + NT - 1) / NT, NT, 0, stream>>>(xout, N_NODES * C_IN);
    lin1_accum<<<MTILES, 64, 0, stream>>>(xout, lin1_w, hid1, kofs);
  };

  gcn(state, W1, b1,  bufA, 0);     // out1
  gcn(bufA,  W2, b2,  bufB, 64);    // out2
  gcn(bufB,  W3, b3g, bufA, 128);   // out3
  gcn(bufA,  W3, b3g, bufB, 192);   // out4 (conv3 reused)
  gcn(bufB,  W3, b3g, bufA, 256);   // out5 (conv3 reused)

  lrelu_ip<<<((size_t)N_NODES * HID + NT - 1) / NT, NT, 0, stream>>>(hid1, N_NODES * HID);
  gemm_lin2<<<MTILES, 64, 0, stream>>>(hid1, lin2_w, lin2_b, hid2);

  zero_small<<<(N_GRAPH + 1 + NT - 1) / NT, NT, 0, stream>>>(gsum, rsum);
  lin3_kernel<<<N_NODES / NT, NT, 0, stream>>>(hid2, lin3_w, lin3_b, out, gsum, rsum, N_NODES);
  normalize_kernel<<<(N_NODES + NT - 1) / NT, NT, 0, stream>>>(out, gsum, rsum, N_NODES);

  (void)in_sizes; (void)n_in; (void)out_size; (void)ws_size;
}